// BSplineLayer_12618613916416
// MI455X (gfx1250) — compile-verified
//
#include <hip/hip_runtime.h>
#include <math.h>

typedef __attribute__((ext_vector_type(2))) float v2f;
typedef __attribute__((ext_vector_type(8))) float v8f;

#define BATCH   256
#define IN_F    128
#define NCP     64
#define NDP     4096
#define NLOGIT  4095
#define DEG     5
#define EPS     1e-7f

// knots: [0]*6, i/59 for i=1..58, [1]*6  (70 entries). Exact f32 rounding via division.
__device__ __forceinline__ float knotf(int j) {
    int jc = j - DEG;
    if (jc < 0) jc = 0;
    if (jc > NCP - DEG) jc = NCP - DEG;   // 59
    return (float)jc / 59.0f;
}

// ---------------- K1: logits = input @ W_lin^T + b_lin via V_WMMA_F32_16X16X4_F32 ----
// Branch-free inner loop: the only out-of-range column (n==4095) is clamped to row
// 4094; its logits column is dead data (softmax kernel only reads n<4095).
__global__ __launch_bounds__(256) void k_gemm_logits(
    const float* __restrict__ A,     // input  [256,128]
    const float* __restrict__ W,     // W_lin  [4095,128]
    const float* __restrict__ bias,  // b_lin  [4095]
    float* __restrict__ logits)      // [256,4096]
{
    const int lane = threadIdx.x & 31;
    const int wave = threadIdx.x >> 5;
    const int tile = blockIdx.x * 8 + wave;     // 16 m-tiles * 256 n-tiles = 4096
    const int mt = tile >> 8;
    const int nt = tile & 255;
    const int m0 = mt * 16, n0 = nt * 16;

    const int r     = lane & 15;
    const int koff  = (lane >> 4) << 1;         // 0 or 2
    const int arow  = m0 + r;
    const int ncol  = n0 + r;
    const int ncolc = (ncol < NLOGIT) ? ncol : (NLOGIT - 1);   // clamp, no masking

    const float2* Ap = (const float2*)(A + (size_t)arow  * IN_F);
    const float2* Wp = (const float2*)(W + (size_t)ncolc * IN_F);

    v8f acc = {};
    #pragma unroll
    for (int k0 = 0; k0 < IN_F; k0 += 4) {
        const int kk = (k0 + koff) >> 1;        // float2 index, k0+koff is even
        const float2 af = Ap[kk];               // A[M=r][k0+koff .. +1]
        const float2 bf = Wp[kk];               // B[k0+koff][N=r] = W_lin[N][k..]
        v2f a; a[0] = af.x; a[1] = af.y;
        v2f b; b[0] = bf.x; b[1] = bf.y;
        acc = __builtin_amdgcn_wmma_f32_16x16x4_f32(
            false, a, false, b, (short)0, acc, false, false);
    }

    const float bl = bias[ncolc];
    // D: lane<16 -> col n0+lane, rows m0+0..7 ; lane>=16 -> col n0+lane-16, rows m0+8..15
    const int rbase = m0 + ((lane >> 4) << 3);
    #pragma unroll
    for (int v = 0; v < 8; ++v)
        logits[(size_t)(rbase + v) * NDP + ncol] = acc[v] + bl;
}

// ---------------- K2: per-row softmax + prefix scan -> ub, intvls ---------------------
__global__ __launch_bounds__(256) void k_softmax_scan(
    const float* __restrict__ logits,   // [256,4096] (col 4095 unused)
    float* __restrict__ ub,             // [256,4096]
    float* __restrict__ intvls)         // [256,4096]
{
    __shared__ float s_v[NDP];          // logits, then exp values (in place)
    __shared__ float s_red[256];
    __shared__ float s_scan[256];
    __shared__ float s_carry;

    const int b   = blockIdx.x;
    const int tid = threadIdx.x;
    const float* lg = logits + (size_t)b * NDP;

    // Stage logits into LDS. Prefer the CDNA5 async global->LDS path when the
    // toolchain exposes it; otherwise plain loads.
#if defined(__gfx1250__) && __has_builtin(__builtin_amdgcn_global_load_async_to_lds_b32)
    for (int n = tid; n < NLOGIT; n += 256) {
        __builtin_amdgcn_global_load_async_to_lds_b32(
            (__attribute__((address_space(1))) int*)(uintptr_t)(lg + n),
            (__attribute__((address_space(3))) int*)(uintptr_t)(&s_v[n]),
            /*imm offset*/0, /*cpol*/0);
    }
    asm volatile("s_wait_asynccnt 0" ::: "memory");
#else
    for (int n = tid; n < NLOGIT; n += 256) s_v[n] = lg[n];
#endif
    __syncthreads();

    // row max
    float m = -__builtin_inff();
    for (int n = tid; n < NLOGIT; n += 256) m = fmaxf(m, s_v[n]);
    s_red[tid] = m; __syncthreads();
    for (int off = 128; off > 0; off >>= 1) {
        if (tid < off) s_red[tid] = fmaxf(s_red[tid], s_red[tid + off]);
        __syncthreads();
    }
    const float mx = s_red[0];
    __syncthreads();

    // exp + row sum (exp values stored back into s_v)
    float s = 0.0f;
    for (int n = tid; n < NLOGIT; n += 256) {
        float e = expf(s_v[n] - mx);
        s_v[n] = e;
        s += e;
    }
    s_red[tid] = s; __syncthreads();
    for (int off = 128; off > 0; off >>= 1) {
        if (tid < off) s_red[tid] += s_red[tid + off];
        __syncthreads();
    }
    const float sum = s_red[0];
    __syncthreads();

    if (tid == 0) s_carry = 0.0f;
    __syncthreads();

    // inclusive scan over intvls = [0, probs[0..4094]], 16 chunks of 256
    for (int c = 0; c < 16; ++c) {
        const int j = c * 256 + tid;
        const float val = (j == 0) ? 0.0f : (s_v[j - 1] / sum);
        s_scan[tid] = val; __syncthreads();
        for (int off = 1; off < 256; off <<= 1) {
            const float t2 = (tid >= off) ? s_scan[tid - off] : 0.0f;
            __syncthreads();
            s_scan[tid] += t2;
            __syncthreads();
        }
        const float carry = s_carry;
        const float incl  = carry + s_scan[tid];
        ub[(size_t)b * NDP + j]     = fminf(fmaxf(incl, 0.0f), 1.0f);
        intvls[(size_t)b * NDP + j] = val;
        __syncthreads();
        if (tid == 0) s_carry = carry + s_scan[255];
        __syncthreads();
    }
}

// ---------------- K3: local degree-5 basis + rational projection ---------------------
__global__ __launch_bounds__(256) void k_basis_project(
    const float* __restrict__ cp,   // control_points [256,3,64]
    const float* __restrict__ w,    // weights        [256,1,64]
    const float* __restrict__ ub,   // [256,4096]
    float* __restrict__ dp)         // [256,3,4096]
{
    __shared__ float s_cpw[3][NCP];
    __shared__ float s_w[NCP];

    const int b   = blockIdx.x;
    const int tid = threadIdx.x;
    if (tid < NCP) s_w[tid] = w[(size_t)b * NCP + tid];
    if (tid < 3 * NCP) {
        const int cidx = tid & 63;
        s_cpw[tid >> 6][cidx] = cp[(size_t)b * 3 * NCP + tid] * w[(size_t)b * NCP + cidx];
    }
    __syncthreads();

    for (int it = 0; it < 16; ++it) {
        const int j = it * 256 + tid;
        const float t = ub[(size_t)b * NDP + j];

        float d0 = 0.0f, d1 = 0.0f, d2 = 0.0f, nw = 0.0f;

        if (t < 1.0f) {             // t==1.0 -> reference basis is identically zero
            int i = (int)floorf(t * 59.0f);
            if (i > 58) i = 58;
            if (i < 0)  i = 0;
            int k = DEG + i;
            if (t < knotf(k) && k > DEG)               --k;   // fp boundary fix-ups
            else if (t >= knotf(k + 1) && k < NCP - 1) ++k;

            float T[12];
            #pragma unroll
            for (int d = 0; d < 12; ++d) T[d] = knotf(k - 5 + d);

            float Nv[6] = {1.0f, 0.0f, 0.0f, 0.0f, 0.0f, 0.0f};
            #pragma unroll
            for (int p = 1; p <= DEG; ++p) {
                #pragma unroll
                for (int r = DEG; r >= 0; --r) {
                    if (r > p) continue;
                    // i2 = k - p + r ; indices into T relative to k-5
                    const float L = (t - T[r - p + 5]) / (T[r + 5] - T[r - p + 5] + EPS);
                    const float R = (T[r + 6] - t)     / (T[r + 6] - T[r - p + 6] + EPS);
                    const float nlo = (r > 0) ? Nv[r - 1] : 0.0f;
                    Nv[r] = L * nlo + R * Nv[r];   // Nv[r] is 0 first time (init)
                }
            }

            #pragma unroll
            for (int r = 0; r < 6; ++r) {
                const int c = k - 5 + r;
                const float nv = Nv[r];
                d0 += nv * s_cpw[0][c];
                d1 += nv * s_cpw[1][c];
                d2 += nv * s_cpw[2][c];
                nw += nv * s_w[c];
            }
        }

        const float den = nw + EPS;
        float* o = dp + (size_t)b * 3 * NDP + j;
        o[0]       = d0 / den;
        o[NDP]     = d1 / den;
        o[2 * NDP] = d2 / den;
    }
}

// ---------------- launcher ------------------------------------------------------------
extern "C" void kernel_launch(void* const* d_in, const int* in_sizes, int n_in,
                              void* d_out, int out_size, void* d_ws, size_t ws_size,
                              hipStream_t stream) {
    const float* input = (const float*)d_in[0];
    const float* cp    = (const float*)d_in[1];
    const float* wts   = (const float*)d_in[2];
    const float* Wlin  = (const float*)d_in[3];
    const float* blin  = (const float*)d_in[4];

    float* out  = (float*)d_out;
    float* dp   = out;                                  // [256,3,4096]
    float* ubp  = out + (size_t)BATCH * 3 * NDP;        // [256,1,4096]
    float* intv = ubp + (size_t)BATCH * NDP;            // [256,4096]

    // logits scratch: prefer d_ws; fall back to dp region (fully overwritten by K3)
    const size_t need = (size_t)BATCH * NDP * sizeof(float);
    float* logits = (ws_size >= need) ? (float*)d_ws : dp;

    k_gemm_logits<<<512, 256, 0, stream>>>(input, Wlin, blin, logits);
    k_softmax_scan<<<BATCH, 256, 0, stream>>>(logits, ubp, intv);
    k_basis_project<<<BATCH, 256, 0, stream>>>(cp, wts, ubp, dp);
}